// Encoder_89558658056521
// MI455X (gfx1250) — compile-verified
//
#include <hip/hip_runtime.h>
#include <stdint.h>

#define N_FLOWS 20
#define WAVE    32
#define NW_BLK  8                    // waves per block -> 256 threads
#define ITEMS   2                    // samples per lane per tile
#define TILE_SAMPLES (WAVE * ITEMS)  // 64 samples (1 KB) per tile
#define TILES_PER_WAVE 8

typedef float v4f __attribute__((ext_vector_type(4)));
typedef int   v4i __attribute__((ext_vector_type(4)));

typedef __attribute__((address_space(1))) v4i* g_v4i_ptr;
typedef __attribute__((address_space(3))) v4i* l_v4i_ptr;

__device__ __forceinline__ float softplus_f(float x) {
  // stable: max(x,0) + log1p(exp(-|x|))
  return fmaxf(x, 0.0f) + log1pf(__expf(-fabsf(x)));
}

// Issue one 16B-per-lane async copy: global -> LDS (CDNA5 ASYNCcnt path).
__device__ __forceinline__ void async_b128_to_lds(const void* gp, const void* lp) {
#if __has_builtin(__builtin_amdgcn_global_load_async_to_lds_b128)
  __builtin_amdgcn_global_load_async_to_lds_b128(
      (g_v4i_ptr)(uintptr_t)gp,
      (l_v4i_ptr)(uint32_t)(uintptr_t)lp,  // flat low 32 bits == LDS offset
      0, 0);
#else
  uint64_t ga = (uint64_t)(uintptr_t)gp;
  uint32_t la = (uint32_t)(uintptr_t)lp;
  asm volatile("global_load_async_to_lds_b128 %0, %1, off"
               :: "v"(la), "v"(ga) : "memory");
#endif
}

__global__ __launch_bounds__(NW_BLK * WAVE) void radial_chain(
    const v4f* __restrict__ X, const float* __restrict__ x0s,
    const float* __restrict__ ap, const float* __restrict__ bp,
    v4f* __restrict__ Y, int nTiles) {
  // wave-private double-buffered stage: 8 waves * 2 bufs * 64 * 16B = 16 KB
  __shared__ v4f stage[NW_BLK][2][TILE_SAMPLES];
  __shared__ alignas(16) float Ps[N_FLOWS * 8];  // {x0[4], alpha, beta, pad[2]}

  const int lane = threadIdx.x & (WAVE - 1);
  const int w    = threadIdx.x >> 5;
  const int wid  = blockIdx.x * NW_BLK + w;
  const int nW   = gridDim.x * NW_BLK;

  // Per-block parameter prep (amortized over thousands of samples/block).
  if (threadIdx.x < N_FLOWS) {
    const int f = threadIdx.x;
    const float a = softplus_f(ap[f]);
    const float b = softplus_f(bp[f]) - a;  // beta = -alpha + softplus(b')
    v4f c;
    c.x = x0s[f * 4 + 0]; c.y = x0s[f * 4 + 1];
    c.z = x0s[f * 4 + 2]; c.w = x0s[f * 4 + 3];
    *(v4f*)&Ps[f * 8] = c;
    Ps[f * 8 + 4] = a;
    Ps[f * 8 + 5] = b;
  }

  // Software-pipeline prologue: prefetch up to 2 tiles for this wave.
  int pend = 0;
  {
    const int t0 = wid;
    if (t0 < nTiles) {
      const v4f* g = X + (size_t)t0 * TILE_SAMPLES;
      async_b128_to_lds(g + lane,        &stage[w][0][lane]);
      async_b128_to_lds(g + WAVE + lane, &stage[w][0][WAVE + lane]);
      pend = 1;
      const int t1 = t0 + nW;
      if (t1 < nTiles) {
        const v4f* g1 = X + (size_t)t1 * TILE_SAMPLES;
        async_b128_to_lds(g1 + lane,        &stage[w][1][lane]);
        async_b128_to_lds(g1 + WAVE + lane, &stage[w][1][WAVE + lane]);
        pend = 2;
      }
    }
  }
  __syncthreads();  // Ps ready for everyone

  int k = 0;
#pragma unroll 1
  for (int t = wid; t < nTiles; t += nW, ++k) {
    // Oldest tile = 2 async instrs; allow the newer tile (2 instrs) in flight.
    if (pend >= 2) { asm volatile("s_wait_asynccnt 0x2" ::: "memory"); }
    else           { asm volatile("s_wait_asynccnt 0x0" ::: "memory"); }
    const int buf = k & 1;
    v4f xa = stage[w][buf][lane];
    v4f xb = stage[w][buf][WAVE + lane];
    pend -= 1;

#pragma unroll
    for (int f = 0; f < N_FLOWS; ++f) {
      const v4f   c  = *(const v4f*)&Ps[f * 8];  // broadcast ds_load_b128
      const float al = Ps[f * 8 + 4];
      const float be = Ps[f * 8 + 5];
      float a0 = xa.x - c.x, a1 = xa.y - c.y, a2 = xa.z - c.z, a3 = xa.w - c.w;
      float b0 = xb.x - c.x, b1 = xb.y - c.y, b2 = xb.z - c.z, b3 = xb.w - c.w;
      float ra2 = fmaf(a3, a3, fmaf(a2, a2, fmaf(a1, a1, a0 * a0)));
      float rb2 = fmaf(b3, b3, fmaf(b2, b2, fmaf(b1, b1, b0 * b0)));
      float ra = __builtin_amdgcn_sqrtf(ra2);
      float rb = __builtin_amdgcn_sqrtf(rb2);
      float sa = be * __builtin_amdgcn_rcpf(al + ra);
      float sb = be * __builtin_amdgcn_rcpf(al + rb);
      xa.x = fmaf(sa, a0, xa.x); xa.y = fmaf(sa, a1, xa.y);
      xa.z = fmaf(sa, a2, xa.z); xa.w = fmaf(sa, a3, xa.w);
      xb.x = fmaf(sb, b0, xb.x); xb.y = fmaf(sb, b1, xb.y);
      xb.z = fmaf(sb, b2, xb.z); xb.w = fmaf(sb, b3, xb.w);
    }

    // Streaming store: data is write-once, bypass-friendly (NT hint).
    v4f* o = Y + (size_t)t * TILE_SAMPLES;
    __builtin_nontemporal_store(xa, o + lane);
    __builtin_nontemporal_store(xb, o + WAVE + lane);

    // Refill this buffer (read at iteration k+2). Ensure our ds reads of the
    // buffer have completed before the async engine overwrites it.
    const int tn = t + 2 * nW;
    if (tn < nTiles) {
      asm volatile("s_wait_dscnt 0x0" ::: "memory");
      const v4f* g = X + (size_t)tn * TILE_SAMPLES;
      async_b128_to_lds(g + lane,        &stage[w][buf][lane]);
      async_b128_to_lds(g + WAVE + lane, &stage[w][buf][WAVE + lane]);
      pend += 1;
    }
  }
}

// Safety tail for nSamples % 64 != 0 (never launched for the reference shape).
__global__ void radial_tail(const float* __restrict__ X, const float* __restrict__ x0s,
                            const float* __restrict__ ap, const float* __restrict__ bp,
                            float* __restrict__ Y, long long base, int cnt) {
  const int i = (int)(threadIdx.x + blockIdx.x * blockDim.x);
  if (i >= cnt) return;
  const long long s = base + i;
  float p0 = X[s * 4 + 0], p1 = X[s * 4 + 1], p2 = X[s * 4 + 2], p3 = X[s * 4 + 3];
  for (int f = 0; f < N_FLOWS; ++f) {
    const float a = softplus_f(ap[f]);
    const float b = softplus_f(bp[f]) - a;
    const float d0 = p0 - x0s[f * 4 + 0];
    const float d1 = p1 - x0s[f * 4 + 1];
    const float d2 = p2 - x0s[f * 4 + 2];
    const float d3 = p3 - x0s[f * 4 + 3];
    const float r  = sqrtf(fmaf(d3, d3, fmaf(d2, d2, fmaf(d1, d1, d0 * d0))));
    const float sc = b / (a + r);
    p0 = fmaf(sc, d0, p0); p1 = fmaf(sc, d1, p1);
    p2 = fmaf(sc, d2, p2); p3 = fmaf(sc, d3, p3);
  }
  Y[s * 4 + 0] = p0; Y[s * 4 + 1] = p1; Y[s * 4 + 2] = p2; Y[s * 4 + 3] = p3;
}

extern "C" void kernel_launch(void* const* d_in, const int* in_sizes, int n_in,
                              void* d_out, int out_size, void* d_ws, size_t ws_size,
                              hipStream_t stream) {
  (void)n_in; (void)out_size; (void)d_ws; (void)ws_size;
  const float* X   = (const float*)d_in[0];
  const float* x0s = (const float*)d_in[1];
  const float* ap  = (const float*)d_in[2];
  const float* bp  = (const float*)d_in[3];
  float* Y = (float*)d_out;

  const long long nSamples = (long long)in_sizes[0] / 4;  // [BATCH, 4]
  const int nTiles = (int)(nSamples / TILE_SAMPLES);
  const int tail   = (int)(nSamples % TILE_SAMPLES);

  if (nTiles > 0) {
    int blocks = (nTiles + NW_BLK * TILES_PER_WAVE - 1) / (NW_BLK * TILES_PER_WAVE);
    radial_chain<<<blocks, NW_BLK * WAVE, 0, stream>>>(
        (const v4f*)X, x0s, ap, bp, (v4f*)Y, nTiles);
  }
  if (tail > 0) {
    radial_tail<<<1, TILE_SAMPLES, 0, stream>>>(
        X, x0s, ap, bp, Y, (long long)nTiles * TILE_SAMPLES, tail);
  }
}